// LightGCN_61632780698011
// MI455X (gfx1250) — compile-verified
//
#include <hip/hip_runtime.h>
#include <hip/hip_bf16.h>
#include <stdint.h>

// ---------------------------------------------------------------------------
// LightGCN propagation on MI455X (gfx1250).
//
// out = (x0 + A x0 + A^2 x0 + A^3 x0) / 4,  A = 9.6M-edge COO, x = [300K, 64].
//
// Roofline: 1.23 GFLOP/layer vs 115 MB/layer edge stream (HBM) + gathers and
// f32 scatter-atomics into node buffers (76.8 MB each -> L2-resident on the
// 192 MB L2).  Bandwidth/atomic-rate bound; WMMA has no productive role
// (random destination rows -> rank-1 scatter).  The CDNA5 features that fit:
//   * global_load_async_to_lds_b32/b128 (ASYNCcnt) to stage edge tiles,
//     with TH=NT so the one-shot edge stream doesn't evict L2-resident x/y
//   * non-returning agent-scope global_atomic_add_f32 for the scatter
// ---------------------------------------------------------------------------

#define DIMS       64       // embedding dim (fixed by reference)
#define TILE_E     256      // edges staged per block
#define BLOCK      256      // 8 waves of 32

// --- CDNA5 async global->LDS path (guarded; falls back cleanly) ------------
#if defined(__has_builtin)
#  if __has_builtin(__builtin_amdgcn_global_load_async_to_lds_b32)
#    define USE_ASYNC_LDS 1
#  endif
#  if __has_builtin(__builtin_amdgcn_global_load_async_to_lds_b128)
#    define HAVE_ASYNC_B128 1
#  endif
#  if __has_builtin(__builtin_amdgcn_s_wait_asynccnt)
#    define HAVE_WAIT_ASYNC 1
#  endif
#endif

#if defined(USE_ASYNC_LDS)
// Probe diagnostics so far:
//   b32 : (addrspace(1) int*, addrspace(3) int*, imm offset, imm cpol)
//   b128: (addrspace(1) v4i*, addrspace(3) v4i*, imm offset, imm cpol)
typedef int v4i __attribute__((vector_size(16)));
typedef __attribute__((address_space(1))) int  g_i32_t;
typedef __attribute__((address_space(3))) int  l_i32_t;
typedef __attribute__((address_space(1))) v4i  g_v4i_t;
typedef __attribute__((address_space(3))) v4i  l_v4i_t;

// gfx12-style CPol: TH in [2:0]; TH=1 -> non-temporal (edge stream is
// read exactly once per layer -- keep it out of the way of resident x/y).
#define EDGE_CPOL 1

__device__ __forceinline__ void async_copy_b32(const void* gptr, void* lptr) {
  __builtin_amdgcn_global_load_async_to_lds_b32(
      (g_i32_t*)gptr, (l_i32_t*)lptr, /*offset=*/0, /*cpol=*/EDGE_CPOL);
}
#if defined(HAVE_ASYNC_B128)
__device__ __forceinline__ void async_copy_b128(const void* gptr, void* lptr) {
  __builtin_amdgcn_global_load_async_to_lds_b128(
      (g_v4i_t*)gptr, (l_v4i_t*)lptr, /*offset=*/0, /*cpol=*/EDGE_CPOL);
}
#endif
__device__ __forceinline__ void async_wait0() {
#if defined(HAVE_WAIT_ASYNC)
  __builtin_amdgcn_s_wait_asynccnt(0);
#else
  asm volatile("s_wait_asynccnt 0" ::: "memory");
#endif
}
#endif  // USE_ASYNC_LDS

// ---------------------------------------------------------------------------
// COO SpMM: y[rows[e]] += vals[e] * x[cols[e]]   (D = 64)
//
// Block = 8 waves.  A 256-edge tile of (row,col,val) is staged into LDS with
// async global->LDS copies (full tile: 192 lanes x one b128 each), then each
// wave owns one edge per iteration: (r,c,v) are wave-uniform LDS broadcasts,
// lane L handles dims [2L, 2L+1] -> one coalesced float2 gather (256 B per
// edge per wave, L2-resident) and two relaxed agent-scope f32 atomics
// (no return -> global_atomic_add_f32 without RET, tracked on STOREcnt).
//
// NOTE: kept first in the file so the --disasm snippet shows this (hot)
// kernel: want to verify global_load_async_to_lds_b128 + s_wait_asynccnt +
// single-instruction global_atomic_add_f32 (no CAS loop).
// ---------------------------------------------------------------------------
__global__ void lgcn_spmm(const float* __restrict__ x,
                          float* __restrict__ y,
                          const float* __restrict__ vals,
                          const int*  __restrict__ rows,
                          const int*  __restrict__ cols,
                          int n_edges) {
  __shared__ int   s_rows[TILE_E];
  __shared__ int   s_cols[TILE_E];
  __shared__ float s_vals[TILE_E];

  const int t    = threadIdx.x;            // 0..255
  const int base = blockIdx.x * TILE_E;
  int count = n_edges - base;
  if (count > TILE_E) count = TILE_E;
  if (count <= 0) return;

  // -------- stage edge tile into LDS --------
#if defined(USE_ASYNC_LDS) && defined(HAVE_ASYNC_B128)
  if (count == TILE_E) {
    // Whole tile: 3 arrays x 64 lanes x 16B async b128 copies.
    if (t < 3 * (TILE_E / 4)) {
      const int arr = t / (TILE_E / 4);    // 0=rows 1=cols 2=vals
      const int idx = (t % (TILE_E / 4)) * 4;
      const void* g;
      void*       l;
      if      (arr == 0) { g = rows + base + idx; l = &s_rows[idx]; }
      else if (arr == 1) { g = cols + base + idx; l = &s_cols[idx]; }
      else               { g = vals + base + idx; l = &s_vals[idx]; }
      async_copy_b128(g, l);
    }
  } else if (t < count) {
    async_copy_b32(rows + base + t, &s_rows[t]);
    async_copy_b32(cols + base + t, &s_cols[t]);
    async_copy_b32(vals + base + t, &s_vals[t]);
  }
  async_wait0();
#elif defined(USE_ASYNC_LDS)
  if (t < count) {
    async_copy_b32(rows + base + t, &s_rows[t]);
    async_copy_b32(cols + base + t, &s_cols[t]);
    async_copy_b32(vals + base + t, &s_vals[t]);
  }
  async_wait0();
#else
  if (t < count) {
    s_rows[t] = __builtin_nontemporal_load(rows + base + t);
    s_cols[t] = __builtin_nontemporal_load(cols + base + t);
    s_vals[t] = __builtin_nontemporal_load(vals + base + t);
  }
#endif
  __syncthreads();

  const int lane = t & 31;                 // dim pair owned by this lane
  const int wave = t >> 5;                 // 0..7 : edge slot stride
  const float2* __restrict__ x2 = (const float2*)x;

#pragma unroll 4
  for (int e = wave; e < count; e += (BLOCK / 32)) {
    const int   r = s_rows[e];             // wave-uniform broadcasts
    const int   c = s_cols[e];
    const float v = s_vals[e];

    float2 xv = x2[(size_t)c * (DIMS / 2) + lane];   // coalesced 256B/edge
    float* yp = y + (size_t)r * DIMS + lane * 2;
    (void)__hip_atomic_fetch_add(yp + 0, v * xv.x,
                                 __ATOMIC_RELAXED, __HIP_MEMORY_SCOPE_AGENT);
    (void)__hip_atomic_fetch_add(yp + 1, v * xv.y,
                                 __ATOMIC_RELAXED, __HIP_MEMORY_SCOPE_AGENT);
  }
}

// ---------------------------------------------------------------------------
// acc = (acc + y) * s   (s = 1 for inner layers, 0.25 fused into last layer)
// ---------------------------------------------------------------------------
__global__ void lgcn_accum(float4* __restrict__ acc4,
                           const float4* __restrict__ y4,
                           float s, int n4) {
  int i = blockIdx.x * blockDim.x + threadIdx.x;
  if (i >= n4) return;
  float4 a = acc4[i];
  float4 b = y4[i];
  a.x = (a.x + b.x) * s;
  a.y = (a.y + b.y) * s;
  a.z = (a.z + b.z) * s;
  a.w = (a.w + b.w) * s;
  acc4[i] = a;
}

// ---------------------------------------------------------------------------
// Host-side orchestration (all graph-capture-safe: async copies/memsets and
// kernel launches on `stream` only).
//   d_in: [user_emb (f32), item_emb (f32), adj_vals (f32), adj_rows (i32),
//          adj_cols (i32)]
//   d_out: [300K, 64] f32 accumulator / final output.
//   d_ws : two ping-pong node buffers (2 * 76.8 MB needed).
// ---------------------------------------------------------------------------
extern "C" void kernel_launch(void* const* d_in, const int* in_sizes, int n_in,
                              void* d_out, int out_size, void* d_ws, size_t ws_size,
                              hipStream_t stream) {
  const float* user = (const float*)d_in[0];
  const float* item = (const float*)d_in[1];
  const float* vals = (const float*)d_in[2];
  const int*   rows = (const int*)  d_in[3];
  const int*   cols = (const int*)  d_in[4];

  const int nU_f = in_sizes[0];            // user floats  (100K * 64)
  const int nI_f = in_sizes[1];            // item floats  (200K * 64)
  const int nE   = in_sizes[2];            // edge count   (9.6M)
  const int nF   = nU_f + nI_f;            // total node floats (19.2M)
  const int n4   = nF / 4;

  float* acc = (float*)d_out;              // running layer sum
  float* xA  = (float*)d_ws;               // current layer input
  float* xB  = xA + nF;                    // current layer output

  // acc = x0, xA = x0  (concat user||item) via DMA copies, not kernels.
  (void)hipMemcpyAsync(acc,        user, (size_t)nU_f * sizeof(float),
                       hipMemcpyDeviceToDevice, stream);
  (void)hipMemcpyAsync(acc + nU_f, item, (size_t)nI_f * sizeof(float),
                       hipMemcpyDeviceToDevice, stream);
  (void)hipMemcpyAsync(xA,         user, (size_t)nU_f * sizeof(float),
                       hipMemcpyDeviceToDevice, stream);
  (void)hipMemcpyAsync(xA + nU_f,  item, (size_t)nI_f * sizeof(float),
                       hipMemcpyDeviceToDevice, stream);

  const int EW_GRID   = (n4 + BLOCK - 1) / BLOCK;
  const int SPMM_GRID = (nE + TILE_E - 1) / TILE_E;

  const int N_LAYERS = 3;
  for (int layer = 0; layer < N_LAYERS; ++layer) {
    (void)hipMemsetAsync(xB, 0, (size_t)nF * sizeof(float), stream);
    lgcn_spmm<<<SPMM_GRID, BLOCK, 0, stream>>>(xA, xB, vals, rows, cols, nE);
    const float s = (layer == N_LAYERS - 1) ? 0.25f : 1.0f;  // fuse /4
    lgcn_accum<<<EW_GRID, BLOCK, 0, stream>>>((float4*)acc, (const float4*)xB,
                                              s, n4);
    float* tmp = xA; xA = xB; xB = tmp;    // ping-pong
  }
}